// ViterbiLoss_88278757802458
// MI455X (gfx1250) — compile-verified
//
#include <hip/hip_runtime.h>
#include <stdint.h>

// Problem constants from the reference: B=64, T=512, K=48
#define KT 48
#define TLEN 512
#define START_TAG 46
#define END_TAG 47
#define NTHR 384                    // 12 wave32s: wave w owns columns 4w..4w+3
#define TILE_ELEMS (KT * KT)        // 2304 floats = 9216 bytes per (b,t) tile
#define TILE_BYTES (TILE_ELEMS * 4)

typedef unsigned int v4u __attribute__((ext_vector_type(4)));
typedef int          v8i __attribute__((ext_vector_type(8)));
typedef int          v4i __attribute__((ext_vector_type(4)));

// Issue one TDM DMA of a contiguous 9216-byte score tile (global -> LDS).
// D# built per CDNA5 ISA 8.3/8.4: 1-D tile, data_size=4B, tile_dim0=2304,
// tensor_dim0=2304 (no OOB), type=2 ("image"), count=1, no multicast/pad.
// This toolchain exposes the 6-arg builtin: (g0 v4u, g1 v8i, v4i, v4i, v8i, cpol).
__device__ __forceinline__ void tdm_load_tile(const float* gsrc, unsigned lds_byte_addr) {
    uint64_t ga = (uint64_t)(uintptr_t)gsrc;
    v4u g0;
    g0[0] = 1u;                                                   // count=1 (valid descriptor)
    g0[1] = lds_byte_addr;                                        // lds_addr [63:32]
    g0[2] = (unsigned)(ga & 0xFFFFFFFFull);                       // global_addr lo
    g0[3] = (unsigned)((ga >> 32) & 0x01FFFFFFull) | 0x80000000u; // global_addr hi, type=2
    v8i g1;
    g1[0] = 0x00020000;                                // workgroup_mask=0, data_size=2 (4B)
    g1[1] = (int)((TILE_ELEMS & 0xFFFF) << 16);        // tensor_dim0[15:0]  @ bits 63:48
    g1[2] = (int)(((TILE_ELEMS >> 16) & 0xFFFF) | (1 << 16)); // dim0 hi, tensor_dim1=1
    g1[3] = (int)((TILE_ELEMS & 0xFFFF) << 16);        // tile_dim0=2304     @ bits 127:112
    g1[4] = 1;                                         // tile_dim1=1, tile_dim2=0
    g1[5] = TILE_ELEMS;                                // tensor_dim0_stride lo (unused for 1D)
    g1[6] = 0;
    g1[7] = 0;
    v4i z4 = {0, 0, 0, 0};
    v8i z8 = {0, 0, 0, 0, 0, 0, 0, 0};
    __builtin_amdgcn_tensor_load_to_lds(g0, g1, z4, z4, z8, 0);
}

__global__ __launch_bounds__(NTHR)
void ViterbiLoss_crf_fwd_kernel(const float* __restrict__ scores,
                                const int* __restrict__ targets,
                                const unsigned char* __restrict__ mask,
                                float* __restrict__ out) {
    __shared__ float s_tile[3][TILE_ELEMS];   // triple-buffered score tiles (27648 B)
    __shared__ float s_alpha[2][KT];          // double-buffered forward variables
    __shared__ float s_M[2];                  // ping-pong log-shift
    __shared__ int   s_tgt[TLEN];
    __shared__ int   s_msk[TLEN];

    const int b    = blockIdx.x;
    const int tid  = threadIdx.x;
    const int lane = tid & 31;
    const int w    = tid >> 5;          // wave id 0..11
    const int sub  = lane & 7;          // i-group within column (8-way K split)
    const int col  = lane >> 3;         // 0..3
    const int j    = (w << 2) + col;    // output column 0..47

    const float* sb = scores + (size_t)b * TLEN * TILE_ELEMS;
    const unsigned lds_tile0 = (unsigned)(uintptr_t)(&s_tile[0][0]);

    // Preload targets & mask for this chain into LDS (off the critical path).
    for (int idx = tid; idx < TLEN; idx += NTHR) {
        s_tgt[idx] = targets[b * TLEN + idx];
        s_msk[idx] = (int)mask[b * TLEN + idx];
    }

    // Scalar-uniform wave id so the TDM issue is a true SALU branch (TDM ignores EXEC).
    const int wid = __builtin_amdgcn_readfirstlane(w);

    if (wid == 0) {
        tdm_load_tile(sb + 0 * (size_t)TILE_ELEMS, lds_tile0 + 0 * TILE_BYTES);
        tdm_load_tile(sb + 1 * (size_t)TILE_ELEMS, lds_tile0 + 1 * TILE_BYTES);
        tdm_load_tile(sb + 2 * (size_t)TILE_ELEMS, lds_tile0 + 2 * TILE_BYTES);
        __builtin_amdgcn_s_wait_tensorcnt(2);   // tile 0 resident (in-order completion)
    }
    __syncthreads();

    float gold = 0.0f;                           // tracked by thread 0 only
    // t = 0: alpha_j = mask ? s0[START_TAG, j] : 0
    if (tid < KT) {
        float v = s_msk[0] ? s_tile[0][START_TAG * KT + tid] : 0.0f;
        s_alpha[1][tid] = v;
        if (tid == 0) {
            s_M[1] = v;                          // log-shift for step 1
            if (s_msk[0]) gold += s_tile[0][s_tgt[0]];
        }
    }
    if (wid == 0) __builtin_amdgcn_s_wait_tensorcnt(1);  // tile 1 resident
    __syncthreads();

    for (int t = 1; t < TLEN; ++t) {
        const float* tile = &s_tile[t % 3][0];
        const int rd = t & 1;
        const int wr = rd ^ 1;
        const float M = s_M[rd];

        // alpha'_j = M + log( sum_i exp(alpha_i - M + s[i][j]) ), 8-way split over i
        float acc = 0.0f;
#pragma unroll
        for (int k = 0; k < 6; ++k) {
            const int i = sub + (k << 3);
            acc += __expf(s_alpha[rd][i] - M + tile[i * KT + j]);
        }
        // combine the 8 partials within the wave (no extra barrier)
        acc += __shfl_xor(acc, 1);
        acc += __shfl_xor(acc, 2);
        acc += __shfl_xor(acc, 4);
        if (sub == 0) {
            float neww = M + __logf(acc);
            float upd  = s_msk[t] ? neww : s_alpha[rd][j];   // masked update
            s_alpha[wr][j] = upd;
            if (j == 0) s_M[wr] = upd;     // next step's shift (within O(1) of max alpha)
        }
        if (tid == 0 && s_msk[t]) gold += tile[s_tgt[t]];

        // Prefetch tile t+2 into the slot tile t-1 vacated; ensure t+1 is resident.
        if (wid == 0) {
            if (t + 2 < TLEN) {
                tdm_load_tile(sb + (size_t)(t + 2) * TILE_ELEMS,
                              lds_tile0 + (unsigned)((t + 2) % 3) * TILE_BYTES);
                __builtin_amdgcn_s_wait_tensorcnt(1);
            } else {
                __builtin_amdgcn_s_wait_tensorcnt(0);
            }
        }
        __syncthreads();   // publishes new alpha/M and tile t+1
    }

    if (tid == 0) {
        out[b] = s_alpha[TLEN & 1][END_TAG] - gold;   // all_paths - gold_score
    }
}

extern "C" void kernel_launch(void* const* d_in, const int* in_sizes, int n_in,
                              void* d_out, int out_size, void* d_ws, size_t ws_size,
                              hipStream_t stream) {
    const float*         scores  = (const float*)d_in[0];
    const int*           targets = (const int*)d_in[1];
    const unsigned char* msk     = (const unsigned char*)d_in[2];  // jnp bool -> 1 byte
    float*               out     = (float*)d_out;

    const int B = out_size;  // 64
    ViterbiLoss_crf_fwd_kernel<<<B, NTHR, 0, stream>>>(scores, targets, msk, out);

    (void)in_sizes; (void)n_in; (void)d_ws; (void)ws_size;
}